// LogSumExpPooling2D_59433757442748
// MI455X (gfx1250) — compile-verified
//
#include <hip/hip_runtime.h>
#include <stdint.h>

// LogSumExp 2x2/stride-2 pooling fused with batch reduction, MI455X (gfx1250).
// Memory-bound: 134MB in / 4MB out -> ~5.9us floor at 23.3 TB/s.
// Data path: TDM (tensor_load_to_lds) DMAs a 64KB [n=8][h=2][16w*64c] tile
// into LDS per workgroup; waves reduce from LDS with v_exp_f32/v_log_f32.

typedef unsigned int v4u __attribute__((ext_vector_type(4)));
typedef int          v4i __attribute__((ext_vector_type(4)));
typedef int          v8i __attribute__((ext_vector_type(8)));

#define N_   8
#define H_   256
#define W_   256
#define C_   64
#define OH_  128
#define OW_  128
#define TWOUT 8                      // output columns per block
#define TILE_X 1024u                 // (2*TWOUT)*C_ contiguous floats (4KB run)
#define TILE_Y 2u                    // h window rows
#define TILE_Z 8u                    // batch

__global__ __launch_bounds__(128)
void lse_pool_tdm_kernel(const float* __restrict__ x, float* __restrict__ out)
{
    // 1024*2*8 floats = 64 KB LDS tile; layout: [(z*2+y)*1024 + x] floats.
    __shared__ float4 tile4[(TILE_X * TILE_Y * TILE_Z) / 4];

    const unsigned blk   = blockIdx.x;        // 2048 blocks = 128 h' * 16 wtiles
    const unsigned hp    = blk >> 4;          // output row 0..127
    const unsigned wt    = blk & 15;          // tile column index 0..15
    const unsigned h_in0 = hp * 2;
    const unsigned w_in0 = wt * (2 * TWOUT);

    // ---- Issue the TDM load from wave 0 only (scalar-uniform branch; TDM
    // ignores EXEC so we want a real s_cbranch around it). -------------------
    if (__builtin_amdgcn_readfirstlane((int)(threadIdx.x >> 5)) == 0) {
        const uint64_t gaddr =
            (uint64_t)(uintptr_t)x +
            ((uint64_t)(h_in0 * W_ + w_in0) * C_) * sizeof(float);
        const uint32_t lds_off = (uint32_t)(uintptr_t)(void*)&tile4[0];

        // D# group 0: count=1, lds_addr, 57b global_addr, type=2 ("image")
        v4u g0;
        g0[0] = 1u;
        g0[1] = lds_off;
        g0[2] = (uint32_t)gaddr;
        g0[3] = ((uint32_t)(gaddr >> 32) & 0x1FFFFFFu) | (2u << 30);

        // strides in data_size (4B) units
        const uint64_t s0 = (uint64_t)(W_ * C_);        // h stride: 16384
        const uint64_t s1 = (uint64_t)(H_ * W_ * C_);   // n stride: 4194304

        // D# group 1: data_size=4B; tensor dims == tile dims (no OOB clip)
        v8i g1;
        g1[0] = (int)(2u << 16);                                   // data_size=2 (4B)
        g1[1] = (int)((TILE_X & 0xFFFFu) << 16);                   // tensor_dim0[15:0]
        g1[2] = (int)(((TILE_X >> 16) & 0xFFFFu) |                 // tensor_dim0[31:16]
                      ((TILE_Y & 0xFFFFu) << 16));                 // tensor_dim1[15:0]
        g1[3] = (int)((TILE_X & 0xFFFFu) << 16);                   // tile_dim0 (tdim1 hi=0)
        g1[4] = (int)(TILE_Y | (TILE_Z << 16));                    // tile_dim1 | tile_dim2
        g1[5] = (int)(uint32_t)s0;                                 // dim0_stride[31:0]
        g1[6] = (int)(((uint32_t)(s0 >> 32) & 0xFFFFu) |           // dim0_stride[47:32]
                      (((uint32_t)s1 & 0xFFFFu) << 16));           // dim1_stride[15:0]
        g1[7] = (int)(uint32_t)(s1 >> 16);                         // dim1_stride[47:16]

        // D# group 2: tensor_dim2 = 8 (z), dims 3/4 unused (tile_dim3 = 0)
        v4i g2; g2[0] = (int)TILE_Z; g2[1] = 0; g2[2] = 0; g2[3] = 0;
        v4i g3 = {0, 0, 0, 0};
        // clang-23 (therock-10.0) 6-arg form carries an extra 8-dword group;
        // unused for a 3-D tile -> zero-filled.
        v8i g4 = {0, 0, 0, 0, 0, 0, 0, 0};

        __builtin_amdgcn_tensor_load_to_lds(g0, g1, g2, g3, g4, /*cpol*/0);
        __builtin_amdgcn_s_wait_tensorcnt(0);
    }
    __syncthreads();

    // ---- Reduce from LDS: each thread does 4 channels of one output col ----
    const int t  = (int)threadIdx.x;   // 0..127
    const int c4 = t & 15;             // float4 channel group
    const int wl = t >> 4;             // output column within tile, 0..7

    const float K   = 100.0f * 1.4426950408889634f;   // SCALE * log2(e)
    const float RLN = 0.6931471805599453f / 100.0f;   // ln2 / SCALE

    float4 vmax = make_float4(-3.402823466e38f, -3.402823466e38f,
                              -3.402823466e38f, -3.402823466e38f);
#pragma unroll
    for (int z = 0; z < 16; ++z) {            // z = n*2 + i
#pragma unroll
        for (int j = 0; j < 2; ++j) {
            float4 v = tile4[z * 256 + (2 * wl + j) * 16 + c4];
            vmax.x = fmaxf(vmax.x, v.x);
            vmax.y = fmaxf(vmax.y, v.y);
            vmax.z = fmaxf(vmax.z, v.z);
            vmax.w = fmaxf(vmax.w, v.w);
        }
    }

    float4 s = make_float4(0.f, 0.f, 0.f, 0.f);
#pragma unroll
    for (int z = 0; z < 16; ++z) {
#pragma unroll
        for (int j = 0; j < 2; ++j) {
            float4 v = tile4[z * 256 + (2 * wl + j) * 16 + c4];
            s.x += __builtin_amdgcn_exp2f((v.x - vmax.x) * K);
            s.y += __builtin_amdgcn_exp2f((v.y - vmax.y) * K);
            s.z += __builtin_amdgcn_exp2f((v.z - vmax.z) * K);
            s.w += __builtin_amdgcn_exp2f((v.w - vmax.w) * K);
        }
    }

    float4 r;
    r.x = vmax.x + __builtin_amdgcn_logf(s.x) * RLN;   // logf == log2 on amdgcn
    r.y = vmax.y + __builtin_amdgcn_logf(s.y) * RLN;
    r.z = vmax.z + __builtin_amdgcn_logf(s.z) * RLN;
    r.w = vmax.w + __builtin_amdgcn_logf(s.w) * RLN;

    ((float4*)out)[(hp * OW_ + wt * TWOUT + wl) * (C_ / 4) + c4] = r;
}

extern "C" void kernel_launch(void* const* d_in, const int* in_sizes, int n_in,
                              void* d_out, int out_size, void* d_ws, size_t ws_size,
                              hipStream_t stream)
{
    (void)in_sizes; (void)n_in; (void)d_ws; (void)ws_size; (void)out_size;
    const float* x = (const float*)d_in[0];
    float* out = (float*)d_out;
    const int blocks = OH_ * (OW_ / TWOUT);   // 128 * 16 = 2048
    lse_pool_tdm_kernel<<<dim3(blocks), dim3(128), 0, stream>>>(x, out);
}